// HeatmapOffsetmapLoss_39917426049306
// MI455X (gfx1250) — compile-verified
//
#include <hip/hip_runtime.h>

typedef __attribute__((ext_vector_type(2))) float v2f;
typedef __attribute__((ext_vector_type(8))) float v8f;

#define B_ 4
#define L_ 19
#define C_ 57          // 3*L
#define H_ 800
#define W_ 640
#define HW_ (H_ * W_)  // 512000
#define R2_ 1600.0f    // 40^2
#define NBL_ (B_ * L_) // 76
#define CHUNK_PX_ 4096
#define CHUNKS_ (HW_ / CHUNK_PX_) // 125, exact

// Workspace layout in floats (all slots written unconditionally; no init pass)
#define WS_BCE_ 0                     // NBL_*CHUNKS_ = 9500 block partials
#define WS_L1X_ (NBL_ * CHUNKS_)      // 76
#define WS_L1Y_ (WS_L1X_ + NBL_)      // 76
#define WS_CNT_ (WS_L1Y_ + NBL_)      // 76

// -------- BCE over the full logits plane, WMMA-accumulated reduction --------
__global__ __launch_bounds__(256) void heatloss_bce_kernel(
    const float* __restrict__ fm, const float* __restrict__ lms,
    float* __restrict__ ws) {
  const int chunk = blockIdx.x;
  const int l = blockIdx.y;
  const int b = blockIdx.z;
  const int bl = b * L_ + l;
  const int tid = threadIdx.x;

  // target point: trunc-to-int like jnp .astype(int32), then back to float
  const float x = (float)(int)(lms[bl * 2 + 0] * (float)H_);
  const float y = (float)(int)(lms[bl * 2 + 1] * (float)W_);

  const float* __restrict__ plane = fm + (size_t)(b * C_ + l) * (size_t)HW_;

  v8f c = {};                       // 16x16 f32 accumulator
  const v2f ones = {1.0f, 1.0f};    // B = ones(4x16)

#pragma unroll
  for (int it = 0; it < 4; ++it) {
    const int p = chunk * CHUNK_PX_ + it * 1024 + tid * 4; // 16B aligned
    const float4 z4 = *(const float4*)(plane + p);
    const int i = p / W_;           // row (W_ divisible by 4 -> shared row)
    const int j = p - i * W_;       // col of first element
    const float dx = x - (float)i;
    const float dx2 = dx * dx;
    const float zz[4] = {z4.x, z4.y, z4.z, z4.w};
    float vv[4];
#pragma unroll
    for (int q = 0; q < 4; ++q) {
      const float z = zz[q];
      const float dy = y - (float)(j + q);
      const float inside = (dx2 + dy * dy <= R2_) ? 1.0f : 0.0f;
      const float e = __expf(-fabsf(z));
      // max(z,0) - z*heat + log1p(exp(-|z|)), 1+e in (1,2] so __logf is fine
      vv[q] = fmaxf(z, 0.0f) - z * inside + __logf(1.0f + e);
    }
    v2f a;
    a.x = vv[0] + vv[2];
    a.y = vv[1] + vv[3];
    // D = A(16x4) * ones(4x16) + C : accumulates 64 lane-values per issue.
    // EXEC is all-ones here (no divergence in this kernel).
    c = __builtin_amdgcn_wmma_f32_16x16x4_f32(
        false, a, false, ones, (short)0, c, false, false);
  }

  // D[m][n] identical across n. Sum VGPRs 0..7 (rows M=r / M=r+8 per half-
  // wave), then fold lane<->lane+16 to get the full 16-row total.
  float t = c[0] + c[1] + c[2] + c[3] + c[4] + c[5] + c[6] + c[7];
  t += __shfl_xor(t, 16, 32);       // wave32

  __shared__ float wsum[8];
  const int wave = tid >> 5;
  if ((tid & 31) == 0) wsum[wave] = t;
  __syncthreads();
  if (tid == 0) {
    float s = 0.0f;
#pragma unroll
    for (int wv = 0; wv < 8; ++wv) s += wsum[wv];
    ws[WS_BCE_ + bl * CHUNKS_ + chunk] = s;   // deterministic: one writer
  }
}

// -------- L1 terms + cnt: only the clipped 81x81 disk bounding box ----------
__global__ __launch_bounds__(256) void heatloss_l1_kernel(
    const float* __restrict__ fm, const float* __restrict__ lms,
    float* __restrict__ ws) {
  const int bl = blockIdx.x;
  const int b = bl / L_;
  const int l = bl - b * L_;

  const int xi = (int)(lms[bl * 2 + 0] * (float)H_);
  const int yi = (int)(lms[bl * 2 + 1] * (float)W_);
  const float x = (float)xi, y = (float)yi;
  const int i0 = max(0, xi - 40), i1 = min(H_ - 1, xi + 40);
  const int j0 = max(0, yi - 40), j1 = min(W_ - 1, yi + 40);
  const int nC = j1 - j0 + 1;
  const int n = (i1 - i0 + 1) * nC;

  const float* __restrict__ px = fm + (size_t)(b * C_ + L_ + l) * (size_t)HW_;
  const float* __restrict__ py = fm + (size_t)(b * C_ + 2 * L_ + l) * (size_t)HW_;

  float lx = 0.0f, ly = 0.0f, cnt = 0.0f;
  for (int cell = threadIdx.x; cell < n; cell += 256) {
    const int r = cell / nC;
    const int i = i0 + r;
    const int j = j0 + (cell - r * nC);
    const float dx = x - (float)i;
    const float dy = y - (float)j;
    if (dx * dx + dy * dy <= R2_) {
      cnt += 1.0f;
      const int off = i * W_ + j;
      lx += fabsf(px[off] - dx * 0.025f);   // 1/40
      ly += fabsf(py[off] - dy * 0.025f);
    }
  }

  __shared__ float s0[256], s1[256], s2[256];
  s0[threadIdx.x] = lx; s1[threadIdx.x] = ly; s2[threadIdx.x] = cnt;
  __syncthreads();
  for (int st = 128; st > 0; st >>= 1) {
    if (threadIdx.x < st) {
      s0[threadIdx.x] += s0[threadIdx.x + st];
      s1[threadIdx.x] += s1[threadIdx.x + st];
      s2[threadIdx.x] += s2[threadIdx.x + st];
    }
    __syncthreads();
  }
  if (threadIdx.x == 0) {
    ws[WS_L1X_ + bl] = s0[0];
    ws[WS_L1Y_ + bl] = s1[0];
    ws[WS_CNT_ + bl] = s2[0];   // >=1 always: center pixel is in bounds
  }
}

// -------- Combine to scalar loss (deterministic sequential sums) ------------
__global__ __launch_bounds__(128) void heatloss_final_kernel(
    const float* __restrict__ ws, float* __restrict__ out) {
  __shared__ float sh[NBL_];
  const int t = threadIdx.x;
  if (t < NBL_) {
    float s = 0.0f;
    for (int ck = 0; ck < CHUNKS_; ++ck) s += ws[WS_BCE_ + t * CHUNKS_ + ck];
    const float bce = s / (float)HW_;
    const float cnt = ws[WS_CNT_ + t];
    sh[t] = 2.0f * bce + ws[WS_L1X_ + t] / cnt + ws[WS_L1Y_ + t] / cnt;
  }
  __syncthreads();
  if (t == 0) {
    float s = 0.0f;
    for (int q = 0; q < NBL_; ++q) s += sh[q];
    out[0] = s / (float)NBL_;
  }
}

extern "C" void kernel_launch(void* const* d_in, const int* in_sizes, int n_in,
                              void* d_out, int out_size, void* d_ws,
                              size_t ws_size, hipStream_t stream) {
  const float* fm = (const float*)d_in[0];   // (4, 57, 800, 640) f32
  const float* lms = (const float*)d_in[1];  // (4, 19, 2) f32
  float* ws = (float*)d_ws;
  float* out = (float*)d_out;

  dim3 gBce(CHUNKS_, L_, B_);
  heatloss_bce_kernel<<<gBce, dim3(256), 0, stream>>>(fm, lms, ws);
  heatloss_l1_kernel<<<dim3(NBL_), dim3(256), 0, stream>>>(fm, lms, ws);
  heatloss_final_kernel<<<dim3(1), dim3(128), 0, stream>>>(ws, out);
}